// HSGPPT_Prompt_20057497272460
// MI455X (gfx1250) — compile-verified
//
#include <hip/hip_runtime.h>
#include <stdint.h>

// ---------------------------------------------------------------------------
// HSGPPT prompt-graph pipeline for MI455X (gfx1250, wave32)
// ---------------------------------------------------------------------------
#define DFEAT 256
#define PPAD  16            // P=10 padded to WMMA M dimension
#define THR_INNER (-1.3862943611f)   // logit(0.2)
#define THR_CROSS (-0.4054651081f)   // logit(0.4)

// Async-LDS staged cross kernel (set to 0 to fall back to direct global loads)
#define USE_ASYNC_STAGE 1

typedef float v2f __attribute__((ext_vector_type(2)));
typedef float v8f __attribute__((ext_vector_type(8)));

__device__ __forceinline__ v8f wmma_f32_16x16x4(v2f a, v2f b, v8f c) {
#if defined(__gfx1250__) && __has_builtin(__builtin_amdgcn_wmma_f32_16x16x4_f32)
  // 8 args: (neg_a, A, neg_b, B, c_mod, C, reuse_a, reuse_b)
  return __builtin_amdgcn_wmma_f32_16x16x4_f32(false, a, false, b, (short)0, c,
                                               false, false);
#else
  c[0] += a[0] * b[0] + a[1] * b[1];
  return c;
#endif
}

// ---------------------------------------------------------------------------
// Phase 1: per-column moments of x (mean, unbiased std)
// ---------------------------------------------------------------------------
__global__ void col_stats_kernel(const float* __restrict__ x, int N,
                                 float* __restrict__ mu_o,
                                 float* __restrict__ sig_o) {
  int col = blockIdx.x;
  float s = 0.f, ss = 0.f;
  for (int r = threadIdx.x; r < N; r += blockDim.x) {
    float v = x[(size_t)r * DFEAT + col];
    s += v;
    ss += v * v;
  }
  __shared__ float sh[256];
  __shared__ float sh2[256];
  sh[threadIdx.x] = s;
  sh2[threadIdx.x] = ss;
  __syncthreads();
  for (int off = 128; off > 0; off >>= 1) {
    if (threadIdx.x < off) {
      sh[threadIdx.x] += sh[threadIdx.x + off];
      sh2[threadIdx.x] += sh2[threadIdx.x + off];
    }
    __syncthreads();
  }
  if (threadIdx.x == 0) {
    float mean = sh[0] / (float)N;
    float var = (sh2[0] - (float)N * mean * mean) / (float)(N - 1);
    mu_o[col] = mean;
    sig_o[col] = sqrtf(fmaxf(var, 0.f)) + 1e-8f;
  }
}

// p_prime = (pf - mu_p)/sig_p * sig_o + mu_o ; write padded [16,256] + tail of x_combined
__global__ void pprime_kernel(const float* __restrict__ pf, int N, int P,
                              const float* __restrict__ mu_o,
                              const float* __restrict__ sig_o,
                              float* __restrict__ ppad,
                              float* __restrict__ xcomb) {
  int d = threadIdx.x;  // 256 threads, one column each
  float s = 0.f, ss = 0.f;
  for (int p = 0; p < P; p++) {
    float v = pf[p * DFEAT + d];
    s += v;
    ss += v * v;
  }
  float mu = s / (float)P;
  float var = (ss - (float)P * mu * mu) / (float)(P - 1);
  float sg = sqrtf(fmaxf(var, 0.f)) + 1e-8f;
  float scale = sig_o[d] / sg;
  for (int p = 0; p < PPAD; p++) {
    float v = (p < P) ? (pf[p * DFEAT + d] - mu) * scale + mu_o[d] : 0.f;
    ppad[p * DFEAT + d] = v;
    if (p < P) xcomb[((size_t)N + p) * DFEAT + d] = v;
  }
}

// ---------------------------------------------------------------------------
// Phase 2: similarity via f32 WMMA (16x16x4), thresholds folded into dot space
// ---------------------------------------------------------------------------
// inner: p_prime @ p_prime^T  (one wave)
__global__ void inner_wmma_kernel(const float* __restrict__ ppad, int P,
                                  unsigned char* __restrict__ innerValid) {
  int lane = threadIdx.x & 31;
  int n = lane & 15, hi = lane >> 4;
  const float* ar = ppad + n * DFEAT + 2 * hi;  // A row == B col == lane%16
  v8f c = {};
  for (int k = 0; k < DFEAT; k += 4) {
    v2f a = *(const v2f*)(ar + k);
    c = wmma_f32_16x16x4(a, a, c);
  }
  for (int r = 0; r < 8; r++) {
    int p = r + 8 * hi;  // C layout: (M = r + 8*(lane/16), N = lane%16)
    int q = n;
    if (p < P && q < P) innerValid[p * P + q] = (c[r] > THR_INNER) ? 1 : 0;
  }
}

// ---------------------------------------------------------------------------
// cross: p_prime @ x^T in 16-row tiles of x.
// Output: per x-row 16-bit prompt mask, stored as two bytes (lo: p0-7, hi: p8-15).
// ---------------------------------------------------------------------------
#if USE_ASYNC_STAGE

#define CROSS_WPB 2                      // waves per block
#define TILE_BYTES (16 * DFEAT * 4)      // one 16-row tile of x = contiguous 16KB

// Stage one contiguous 16KB tile (rows stage_row..stage_row+15) into LDS with
// 32 coalesced async b128 transfers per wave (512B per wave-instruction).
__device__ __forceinline__ void async_stage_tile(const float* __restrict__ x,
                                                 long long stage_row,
                                                 unsigned lds_base, int lane) {
  unsigned long long g = (unsigned long long)(uintptr_t)x +
                         (unsigned long long)stage_row * (DFEAT * 4) +
                         (unsigned)lane * 16u;
  unsigned l = lds_base + (unsigned)lane * 16u;
#pragma unroll
  for (int i = 0; i < TILE_BYTES / 512; ++i) {
    asm volatile("global_load_async_to_lds_b128 %0, %1, off"
                 :
                 : "v"(l), "v"(g)
                 : "memory");
    l += 512u;
    g += 512ull;
  }
}

__global__ void cross_wmma_kernel(const float* __restrict__ x,
                                  const float* __restrict__ ppad, int N, int P,
                                  unsigned char* __restrict__ cv) {
  __shared__ float sx[2 * CROSS_WPB][16 * DFEAT];  // 2 waves x double buffer = 64KB
  const int wave = threadIdx.x >> 5;
  const int lane = threadIdx.x & 31;
  const int n = lane & 15, hi = lane >> 4;
  const int ntiles = (N + 15) >> 4;
  const int nwaves = gridDim.x * CROSS_WPB;
  const int wid = blockIdx.x * CROSS_WPB + wave;

  if (wid >= ntiles) return;  // wave-uniform: EXEC stays all-ones for WMMA

  const float* ar = ppad + n * DFEAT + 2 * hi;  // A fragment (L2-resident 16KB)
  unsigned ldsoff[2];
  ldsoff[0] = (unsigned)(uintptr_t)(&sx[2 * wave + 0][0]);
  ldsoff[1] = (unsigned)(uintptr_t)(&sx[2 * wave + 1][0]);

  // stage_row helper: clamp so the 16KB copy never reads past the end of x
  long long srow0 = (long long)wid * 16;
  if (srow0 + 16 > N) srow0 = (N > 16) ? (N - 16) : 0;
  async_stage_tile(x, srow0, ldsoff[0], lane);

  int buf = 0;
  long long cur_srow = srow0;
  for (int tile = wid; tile < ntiles; tile += nwaves, buf ^= 1) {
    // prefetch next tile into the other buffer (clamped => always 32 issues)
    int nxt = (tile + nwaves < ntiles) ? (tile + nwaves) : tile;
    long long nsrow = (long long)nxt * 16;
    if (nsrow + 16 > N) nsrow = (N > 16) ? (N - 16) : 0;
    async_stage_tile(x, nsrow, ldsoff[buf ^ 1], lane);

    // in-order async completion: <=32 outstanding => current buffer resident
    asm volatile("s_wait_asynccnt 0x20" ::: "memory");

    const float* tb = &sx[2 * wave + buf][0];
    int row = tile * 16 + n;                       // x row this lane's B column holds
    int local = (int)((long long)row - cur_srow);  // row within staged tile
    int localc = (local < 0) ? 0 : ((local > 15) ? 15 : local);
    const float* br = tb + localc * DFEAT + 2 * hi;

    v8f c = {};
    for (int k = 0; k < DFEAT; k += 4) {
      v2f a = *(const v2f*)(ar + k);       // global (L2)
      v2f b = *(const v2f*)(br + k);       // LDS (ds_load_b64)
      c = wmma_f32_16x16x4(a, b, c);
    }
    unsigned int m = 0;
    for (int r = 0; r < 8; r++) {
      int p = r + 8 * hi;
      if (p < P && c[r] > THR_CROSS) m |= 1u << r;
    }
    if (row < N) cv[2 * (size_t)row + hi] = (unsigned char)m;
    cur_srow = nsrow;
  }
  // leftover prefetch is drained by S_ENDPGM's implicit wait-idle
}

#else  // direct-global-load fallback

#define CROSS_WPB 8
__global__ void cross_wmma_kernel(const float* __restrict__ x,
                                  const float* __restrict__ ppad, int N, int P,
                                  unsigned char* __restrict__ cv) {
  __shared__ float sp[PPAD * DFEAT];
  for (int i = threadIdx.x; i < PPAD * DFEAT; i += blockDim.x) sp[i] = ppad[i];
  __syncthreads();
  int wave = threadIdx.x >> 5;
  int lane = threadIdx.x & 31;
  int tile = blockIdx.x * CROSS_WPB + wave;
  int ntiles = (N + 15) >> 4;
  if (tile >= ntiles) return;
  int n = lane & 15, hi = lane >> 4;
  int row = tile * 16 + n;
  int rowc = (row < N) ? row : (N - 1);
  const float* ar = sp + n * DFEAT + 2 * hi;
  const float* br = x + (size_t)rowc * DFEAT + 2 * hi;
  v8f c = {};
  for (int k = 0; k < DFEAT; k += 4) {
    v2f a = *(const v2f*)(ar + k);
    v2f b = *(const v2f*)(br + k);
    c = wmma_f32_16x16x4(a, b, c);
  }
  unsigned int m = 0;
  for (int r = 0; r < 8; r++) {
    int p = r + 8 * hi;
    if (p < P && c[r] > THR_CROSS) m |= 1u << r;
  }
  if (row < N) cv[2 * (size_t)row + hi] = (unsigned char)m;
}

#endif  // USE_ASYNC_STAGE

// ---------------------------------------------------------------------------
// Generic fills
// ---------------------------------------------------------------------------
__global__ void fill_u32_kernel(unsigned int* p, unsigned int v, long long n) {
  long long i = (long long)blockIdx.x * blockDim.x + threadIdx.x;
  if (i < n) p[i] = v;
}
__global__ void fill_i32_kernel(int* p, int v, long long n) {
  long long i = (long long)blockIdx.x * blockDim.x + threadIdx.x;
  if (i < n) p[i] = v;
}
__global__ void fill_f32_kernel(float* p, float v, long long n) {
  long long i = (long long)blockIdx.x * blockDim.x + threadIdx.x;
  if (i < n) p[i] = v;
}

// ---------------------------------------------------------------------------
// Phase 3a: counting sort of original edges by src
// ---------------------------------------------------------------------------
__global__ void count_src_kernel(const int* __restrict__ src, int E,
                                 unsigned int* __restrict__ cnt) {
  int i = blockIdx.x * blockDim.x + threadIdx.x;
  if (i < E) atomicAdd(&cnt[src[i]], 1u);
}

__global__ void scatter_kernel(const int* __restrict__ src,
                               const int* __restrict__ dst, int E,
                               const unsigned int* __restrict__ srcStart,
                               unsigned int* __restrict__ cursor,
                               unsigned int* __restrict__ dstbuf) {
  int i = blockIdx.x * blockDim.x + threadIdx.x;
  if (i >= E) return;
  int s = src[i];
  unsigned int pos = srcStart[s] + atomicAdd(&cursor[s], 1u);
  dstbuf[pos] = (unsigned int)dst[i];
}

// per-src segment: sort dsts ascending + unique in place; mean length E/N ~ 16
__global__ void segsort_kernel(unsigned int* __restrict__ dstbuf,
                               const unsigned int* __restrict__ srcStart,
                               const unsigned int* __restrict__ cnt,
                               unsigned int* __restrict__ m_orig, int N) {
  int j = blockIdx.x * blockDim.x + threadIdx.x;
  if (j >= N) return;
  unsigned int* seg = dstbuf + srcStart[j];
  int c = (int)cnt[j];
  for (int i = 1; i < c; i++) {
    unsigned int key = seg[i];
    int k = i - 1;
    while (k >= 0 && seg[k] > key) {
      seg[k + 1] = seg[k];
      k--;
    }
    seg[k + 1] = key;
  }
  int u = 0;
  for (int i = 0; i < c; i++) {
    if (u == 0 || seg[i] != seg[u - 1]) seg[u++] = seg[i];
  }
  m_orig[j] = (unsigned int)u;
}

// ---------------------------------------------------------------------------
// Reusable exclusive scan (n <= 1024*1024)
// ---------------------------------------------------------------------------
#define SCAN_BLOCK 256
#define SCAN_ITEMS 4
#define SCAN_TILE (SCAN_BLOCK * SCAN_ITEMS)

__global__ void scan_blocks_kernel(const unsigned int* __restrict__ in,
                                   unsigned int* __restrict__ out,
                                   unsigned int* __restrict__ partials, int n) {
  __shared__ unsigned int s[SCAN_BLOCK];
  int t = threadIdx.x;
  int base = blockIdx.x * SCAN_TILE + t * SCAN_ITEMS;
  unsigned int v[SCAN_ITEMS];
  unsigned int tsum = 0;
  for (int i = 0; i < SCAN_ITEMS; i++) {
    int idx = base + i;
    v[i] = (idx < n) ? in[idx] : 0u;
    tsum += v[i];
  }
  s[t] = tsum;
  __syncthreads();
  for (int off = 1; off < SCAN_BLOCK; off <<= 1) {
    unsigned int tv = (t >= off) ? s[t - off] : 0u;
    __syncthreads();
    s[t] += tv;
    __syncthreads();
  }
  unsigned int run = s[t] - tsum;  // exclusive prefix of this thread
  for (int i = 0; i < SCAN_ITEMS; i++) {
    int idx = base + i;
    if (idx < n) out[idx] = run;
    run += v[i];
  }
  if (t == SCAN_BLOCK - 1) partials[blockIdx.x] = s[SCAN_BLOCK - 1];
}

__global__ void scan_partials_kernel(unsigned int* __restrict__ partials, int np) {
  __shared__ unsigned int s[1024];
  int t = threadIdx.x;
  unsigned int v = (t < np) ? partials[t] : 0u;
  s[t] = v;
  __syncthreads();
  for (int off = 1; off < 1024; off <<= 1) {
    unsigned int tv = (t >= off) ? s[t - off] : 0u;
    __syncthreads();
    s[t] += tv;
    __syncthreads();
  }
  if (t < np) partials[t] = s[t] - v;  // exclusive
}

__global__ void scan_add_kernel(unsigned int* __restrict__ out,
                                const unsigned int* __restrict__ partials, int n) {
  int i = blockIdx.x * blockDim.x + threadIdx.x;
  if (i < n) out[i] += partials[i / SCAN_TILE];
}

static void scan_exclusive(const unsigned int* in, unsigned int* out, int n,
                           unsigned int* partials, hipStream_t stream) {
  int nb = (n + SCAN_TILE - 1) / SCAN_TILE;  // must be <= 1024
  scan_blocks_kernel<<<nb, SCAN_BLOCK, 0, stream>>>(in, out, partials, n);
  scan_partials_kernel<<<1, 1024, 0, stream>>>(partials, nb);
  scan_add_kernel<<<(n + 255) / 256, 256, 0, stream>>>(out, partials, n);
}

// ---------------------------------------------------------------------------
// Phase 3b: layout counts + emission
// ---------------------------------------------------------------------------
__global__ void bits_kernel(const unsigned char* __restrict__ cv, int N, int P,
                            unsigned int* __restrict__ bits) {
  long long idx = (long long)blockIdx.x * blockDim.x + threadIdx.x;
  if (idx >= (long long)P * N) return;
  int p = (int)(idx / N), j = (int)(idx % N);
  unsigned char byte = cv[2 * (size_t)j + (p >> 3)];
  bits[idx] = (byte >> (p & 7)) & 1u;
}

__global__ void finalize_counts_kernel(const unsigned int* __restrict__ bits,
                                       const unsigned int* __restrict__ bitScan,
                                       const unsigned char* __restrict__ innerValid,
                                       int N, int P, unsigned int* __restrict__ Cp,
                                       unsigned int* __restrict__ innerCnt) {
  int p = threadIdx.x;
  if (p >= P) return;
  size_t last = (size_t)p * N + N - 1;
  Cp[p] = bitScan[last] + bits[last];
  unsigned int ic = 0;
  for (int q = 0; q < P; q++) ic += innerValid[p * P + q];
  innerCnt[p] = ic;
}

__global__ void tot_kernel(const unsigned int* __restrict__ m_orig,
                           const unsigned char* __restrict__ cv,
                           const unsigned int* __restrict__ Cp,
                           const unsigned int* __restrict__ innerCnt, int N, int P,
                           unsigned int* __restrict__ tot) {
  int i = blockIdx.x * blockDim.x + threadIdx.x;
  int T = N + P;
  if (i >= T) return;
  if (i < N) {
    unsigned int mask =
        (unsigned int)cv[2 * (size_t)i] | ((unsigned int)cv[2 * (size_t)i + 1] << 8);
    tot[i] = m_orig[i] + __popc(mask & ((1u << P) - 1u));
  } else {
    int p = i - N;
    tot[i] = Cp[p] + innerCnt[p];
  }
}

// src j < N: unique sorted originals, then reversed-cross (j, N+p) p ascending
__global__ void emit_orig_crossrev_kernel(
    const unsigned int* __restrict__ dstbuf, const unsigned int* __restrict__ srcStart,
    const unsigned int* __restrict__ m_orig, const unsigned int* __restrict__ outStart,
    const unsigned char* __restrict__ cv, int N, int P, int* __restrict__ esrc,
    int* __restrict__ edst, float* __restrict__ ew) {
  int j = blockIdx.x * blockDim.x + threadIdx.x;
  if (j >= N) return;
  unsigned int base = outStart[j];
  unsigned int sb = srcStart[j];
  int m = (int)m_orig[j];
  for (int i = 0; i < m; i++) {
    esrc[base + i] = j;
    edst[base + i] = (int)dstbuf[sb + i];
    ew[base + i] = 1.f;
  }
  unsigned int mask =
      ((unsigned int)cv[2 * (size_t)j] | ((unsigned int)cv[2 * (size_t)j + 1] << 8)) &
      ((1u << P) - 1u);
  unsigned int o = base + m;
  while (mask) {
    int p = __ffs(mask) - 1;
    mask &= mask - 1;
    esrc[o] = j;
    edst[o] = N + p;
    ew[o] = 1.f;
    o++;
  }
}

// src N+p: cross edges (N+p, j), j ascending via per-p bit scan
__global__ void emit_cross_kernel(const unsigned int* __restrict__ bits,
                                  const unsigned int* __restrict__ bitScan,
                                  const unsigned int* __restrict__ outStart, int N,
                                  int P, int* __restrict__ esrc,
                                  int* __restrict__ edst, float* __restrict__ ew) {
  long long idx = (long long)blockIdx.x * blockDim.x + threadIdx.x;
  if (idx >= (long long)P * N) return;
  if (!bits[idx]) return;
  int p = (int)(idx / N), j = (int)(idx % N);
  unsigned int pos = outStart[N + p] + bitScan[idx];
  esrc[pos] = N + p;
  edst[pos] = j;
  ew[pos] = 1.f;
}

// src N+p: inner edges (N+p, N+q) after the C(p) cross edges
__global__ void emit_inner_kernel(const unsigned char* __restrict__ innerValid,
                                  const unsigned int* __restrict__ outStart,
                                  const unsigned int* __restrict__ Cp, int N, int P,
                                  int* __restrict__ esrc, int* __restrict__ edst,
                                  float* __restrict__ ew) {
  int p = threadIdx.x;
  if (p >= P) return;
  unsigned int pos = outStart[N + p] + Cp[p];
  for (int q = 0; q < P; q++) {
    if (innerValid[p * P + q]) {
      esrc[pos] = N + p;
      edst[pos] = N + q;
      ew[pos] = 1.f;
      pos++;
    }
  }
}

// ---------------------------------------------------------------------------
// Host driver (graph-capture safe: only kernels + hipMemcpyAsync on stream)
// ---------------------------------------------------------------------------
extern "C" void kernel_launch(void* const* d_in, const int* in_sizes, int n_in,
                              void* d_out, int out_size, void* d_ws, size_t ws_size,
                              hipStream_t stream) {
  const float* x = (const float*)d_in[0];
  const int* ei = (const int*)d_in[1];
  const float* pf = (const float*)d_in[2];

  const int N = in_sizes[0] / DFEAT;
  const int E = in_sizes[1] / 2;
  const int P = in_sizes[2] / DFEAT;
  const int T = N + P;
  const long long Mmax = (long long)E + (long long)P * P + 2ll * P * N;

  const int* e_src = ei;
  const int* e_dst = ei + E;

  float* xcomb = (float*)d_out;
  int* esrc = (int*)(xcomb + (size_t)T * DFEAT);
  int* edst = esrc + (size_t)Mmax;
  float* ew = (float*)(edst + (size_t)Mmax);

  // --- carve workspace
  char* w = (char*)d_ws;
  auto carve = [&](size_t bytes) -> void* {
    void* p = (void*)w;
    w += (bytes + 255) & ~(size_t)255;
    return p;
  };
  float* mu_o = (float*)carve(DFEAT * 4);
  float* sig_o = (float*)carve(DFEAT * 4);
  float* ppad = (float*)carve((size_t)PPAD * DFEAT * 4);
  unsigned char* cv = (unsigned char*)carve(2 * (size_t)N);
  unsigned char* innerValid = (unsigned char*)carve((size_t)P * P);
  unsigned int* innerCnt = (unsigned int*)carve((size_t)P * 4);
  unsigned int* Cp = (unsigned int*)carve((size_t)P * 4);
  unsigned int* cnt = (unsigned int*)carve((size_t)T * 4);
  unsigned int* cursor = (unsigned int*)carve((size_t)T * 4);
  unsigned int* srcStart = (unsigned int*)carve((size_t)T * 4);
  unsigned int* dstbuf = (unsigned int*)carve((size_t)E * 4);
  unsigned int* m_orig = (unsigned int*)carve((size_t)N * 4);
  unsigned int* tot = (unsigned int*)carve((size_t)T * 4);
  unsigned int* outStart = (unsigned int*)carve((size_t)T * 4);
  unsigned int* bits = (unsigned int*)carve((size_t)P * N * 4);
  unsigned int* bitScan = (unsigned int*)carve((size_t)P * N * 4);
  unsigned int* partials = (unsigned int*)carve(1024 * 4);

  // --- Phase 1: moments + p_prime + x_combined
  col_stats_kernel<<<DFEAT, 256, 0, stream>>>(x, N, mu_o, sig_o);
  pprime_kernel<<<1, DFEAT, 0, stream>>>(pf, N, P, mu_o, sig_o, ppad, xcomb);
  hipMemcpyAsync(xcomb, x, (size_t)N * DFEAT * sizeof(float),
                 hipMemcpyDeviceToDevice, stream);

  // --- Phase 2: WMMA similarities
  inner_wmma_kernel<<<1, 32, 0, stream>>>(ppad, P, innerValid);
  {
    int ntiles = (N + 15) >> 4;
#if USE_ASYNC_STAGE
    // ~4 tiles per wave so the double-buffered async pipeline has depth
    int blocks = (ntiles + CROSS_WPB * 4 - 1) / (CROSS_WPB * 4);
    if (blocks < 1) blocks = 1;
#else
    int blocks = (ntiles + CROSS_WPB - 1) / CROSS_WPB;
#endif
    cross_wmma_kernel<<<blocks, CROSS_WPB * 32, 0, stream>>>(x, ppad, N, P, cv);
  }

  // --- Phase 3a: counting sort of original edges by src + per-src unique
  fill_u32_kernel<<<(T + 255) / 256, 256, 0, stream>>>(cnt, 0u, T);
  fill_u32_kernel<<<(T + 255) / 256, 256, 0, stream>>>(cursor, 0u, T);
  count_src_kernel<<<(E + 255) / 256, 256, 0, stream>>>(e_src, E, cnt);
  scan_exclusive(cnt, srcStart, T, partials, stream);
  scatter_kernel<<<(E + 255) / 256, 256, 0, stream>>>(e_src, e_dst, E, srcStart,
                                                      cursor, dstbuf);
  segsort_kernel<<<(N + 255) / 256, 256, 0, stream>>>(dstbuf, srcStart, cnt, m_orig,
                                                      N);

  // --- Phase 3b: per-prompt bit scans and output offsets
  {
    long long PN = (long long)P * N;
    bits_kernel<<<(int)((PN + 255) / 256), 256, 0, stream>>>(cv, N, P, bits);
    for (int p = 0; p < P; p++)
      scan_exclusive(bits + (size_t)p * N, bitScan + (size_t)p * N, N, partials,
                     stream);
  }
  finalize_counts_kernel<<<1, 32, 0, stream>>>(bits, bitScan, innerValid, N, P, Cp,
                                               innerCnt);
  tot_kernel<<<(T + 255) / 256, 256, 0, stream>>>(m_orig, cv, Cp, innerCnt, N, P,
                                                  tot);
  scan_exclusive(tot, outStart, T, partials, stream);

  // --- emission: prefill with sentinel T / weight 0, then overwrite valid slots
  {
    int gb = (int)((Mmax + 255) / 256);
    fill_i32_kernel<<<gb, 256, 0, stream>>>(esrc, T, Mmax);
    fill_i32_kernel<<<gb, 256, 0, stream>>>(edst, T, Mmax);
    fill_f32_kernel<<<gb, 256, 0, stream>>>(ew, 0.f, Mmax);
  }
  emit_orig_crossrev_kernel<<<(N + 255) / 256, 256, 0, stream>>>(
      dstbuf, srcStart, m_orig, outStart, cv, N, P, esrc, edst, ew);
  {
    long long PN = (long long)P * N;
    emit_cross_kernel<<<(int)((PN + 255) / 256), 256, 0, stream>>>(
        bits, bitScan, outStart, N, P, esrc, edst, ew);
  }
  emit_inner_kernel<<<1, 32, 0, stream>>>(innerValid, outStart, Cp, N, P, esrc, edst,
                                          ew);
}